// GNN_2465311228180
// MI455X (gfx1250) — compile-verified
//
#include <hip/hip_runtime.h>

#define N_NODES 100000
#define N_EDGES 1000000
#define HID     64
#define NGRAPH  64
#define NOUT    2

typedef __attribute__((ext_vector_type(2))) float v2f;
typedef __attribute__((ext_vector_type(8))) float v8f;

// ---------------------------------------------------------------- utilities

__global__ void zero_kernel(float* __restrict__ p, long n) {
    long i = (long)blockIdx.x * blockDim.x + threadIdx.x;
    if (i < n) p[i] = 0.0f;
}

// in-degree count: count[dst]++ per edge (int atomics, done once per call)
__global__ void count_kernel(const int* __restrict__ dst, int* __restrict__ count) {
    long e = (long)blockIdx.x * blockDim.x + threadIdx.x;
    if (e < N_EDGES) atomicAdd(&count[dst[e]], 1);
}

// Single-workgroup chunked exclusive scan of count[] -> offsets[], cursor[].
// Also emits dinv[i] = rsqrt(in_deg + 1) (the +1 is the self loop).
__global__ __launch_bounds__(1024) void scan_kernel(const int* __restrict__ count,
                                                    int* __restrict__ offsets,
                                                    int* __restrict__ cursor,
                                                    float* __restrict__ dinv) {
    __shared__ int tmp[1024];
    __shared__ int carry;
    const int tid = threadIdx.x;
    if (tid == 0) carry = 0;
    __syncthreads();

    for (long base = 0; base < N_NODES; base += 1024) {
        const long i = base + tid;
        int v = (i < N_NODES) ? count[i] : 0;
        if (i < N_NODES) dinv[i] = rsqrtf((float)v + 1.0f);
        tmp[tid] = v;
        __syncthreads();
        // Hillis-Steele inclusive scan
        for (int off = 1; off < 1024; off <<= 1) {
            int t = (tid >= off) ? tmp[tid - off] : 0;
            __syncthreads();
            tmp[tid] += t;
            __syncthreads();
        }
        const int excl = carry + tmp[tid] - v;
        if (i < N_NODES) { offsets[i] = excl; cursor[i] = excl; }
        __syncthreads();
        if (tid == 1023) carry += tmp[1023];
        __syncthreads();
    }
    if (tid == 0) offsets[N_NODES] = carry;   // == N_EDGES
}

// scatter edges into CSR slots: csr_src[cursor[dst]++] = src
__global__ void fill_kernel(const int* __restrict__ src, const int* __restrict__ dst,
                            int* __restrict__ cursor, int* __restrict__ csr_src) {
    long e = (long)blockIdx.x * blockDim.x + threadIdx.x;
    if (e < N_EDGES) {
        const int d = dst[e];
        const int pos = atomicAdd(&cursor[d], 1);
        csr_src[pos] = src[e];
    }
}

// ------------------------------------------------------- WMMA f32 GEMM
// Hout[N,64] = A[N,64] @ W[64,64], exact f32 via V_WMMA_F32_16X16X4_F32.
// One wave computes one 16x16 tile; 4 waves/block cover 16 rows x 64 cols.
// Tile staging uses CDNA5 async global->LDS copies (ASYNCcnt-tracked).
__global__ __launch_bounds__(128) void gemm_wmma_kernel(const float* __restrict__ A,
                                                        const float* __restrict__ W,
                                                        float* __restrict__ Hout) {
    __shared__ float sW[HID * HID];   // 16 KB
    __shared__ float sA[16 * HID];    // 4 KB

    const int tid  = threadIdx.x;     // 0..127
    const int wave = tid >> 5;        // 0..3 -> column tile
    const int lane = tid & 31;
    const long rowBase = (long)blockIdx.x * 16;

    // Async-stage W (64x64) and the 16x64 A tile into LDS (B128 per lane).
    {
        const unsigned sWoff = (unsigned)(size_t)sW;  // LDS byte offset = low 32 bits
        const unsigned sAoff = (unsigned)(size_t)sA;
        const float4* W4 = (const float4*)W;
        const float4* A4 = (const float4*)(A + rowBase * HID);
        #pragma unroll
        for (int i = tid; i < (HID * HID / 4); i += 128) {
            const float4* gp = W4 + i;
            const unsigned lo = sWoff + i * 16;
            asm volatile("global_load_async_to_lds_b128 %0, %1, off"
                         :: "v"(lo), "v"(gp) : "memory");
        }
        if (tid < (16 * HID / 4) - 96) { } // no-op; A tile handled below
        if (tid < 32) {
            const float4* gp = A4 + tid;
            const unsigned lo = sAoff + tid * 16;
            asm volatile("global_load_async_to_lds_b128 %0, %1, off"
                         :: "v"(lo), "v"(gp) : "memory");
        }
        asm volatile("s_wait_asynccnt 0" ::: "memory");
    }
    __syncthreads();

    // 32-bit 16x4 A layout: lanes 0-15 hold M=lane {K=k0,k0+1}; lanes 16-31 {K=k0+2,k0+3}.
    const int m     = lane & 15;
    const int koff  = (lane >> 4) * 2;     // 0 or 2
    const int ncol  = lane & 15;
    const int nbase = wave * 16;

    v8f c = {};
    #pragma unroll
    for (int kk = 0; kk < HID / 4; ++kk) {
        const int k0 = kk * 4 + koff;
        v2f a, b;
        a.x = sA[m * HID + k0];
        a.y = sA[m * HID + k0 + 1];
        b.x = sW[k0 * HID + nbase + ncol];
        b.y = sW[(k0 + 1) * HID + nbase + ncol];
        c = __builtin_amdgcn_wmma_f32_16x16x4_f32(
                false, a, false, b, (short)0, c, false, false);
    }

    // C/D layout: VGPR v -> row (v + 8*(lane>>4)), col (lane&15).
    float* out = Hout + rowBase * HID + nbase;
    const int rhalf = (lane >> 4) * 8;
    #pragma unroll
    for (int v = 0; v < 8; ++v)
        out[(long)(v + rhalf) * HID + ncol] = c[v];
}

// --------------------------------------------- fused gather + self + bias + relu
// Half-wave (16 lanes, float4 each) per dst node; CSR in-edges; zero atomics.
// out[i] = relu( sum_{s in N(i)} h[s]*dinv[s]*dinv[i] + h[i]*dinv[i]^2 + bias )
__global__ __launch_bounds__(256) void gather_kernel(const int* __restrict__ offsets,
                                                     const int* __restrict__ csr_src,
                                                     const float* __restrict__ dinv,
                                                     const float* __restrict__ h,
                                                     const float* __restrict__ bias,
                                                     float* __restrict__ out) {
    const long t = (long)blockIdx.x * 256 + threadIdx.x;
    const long i = t >> 4;                  // node (exactly N_NODES = 6250*16)
    const int  j = (int)(t & 15);           // float4 slot
    const int  lane     = threadIdx.x & 31;
    const int  lanebase = lane & 16;        // base lane of this half-wave
    const int  lsub     = lane & 15;

    const float di = dinv[i];
    const int beg = offsets[i];
    const int end = offsets[i + 1];

    // self-loop term + bias
    const float4 hs = ((const float4*)(h + i * HID))[j];
    const float4 bb = ((const float4*)bias)[j];
    const float d2 = di * di;
    float4 acc;
    acc.x = hs.x * d2 + bb.x;
    acc.y = hs.y * d2 + bb.y;
    acc.z = hs.z * d2 + bb.z;
    acc.w = hs.w * d2 + bb.w;

    // 16 edges per chunk: each lane loads one (src, norm), then broadcast via shfl
    for (int e0 = beg; e0 < end; e0 += 16) {
        const int ee = e0 + lsub;
        int   sv = 0;
        float nv = 0.0f;
        if (ee < end) {
            sv = csr_src[ee];
            nv = dinv[sv] * di;
        }
        const int rem = end - e0;
        const int cnt = rem < 16 ? rem : 16;
        for (int k = 0; k < cnt; ++k) {
            const int   s   = __shfl(sv, lanebase + k, 32);
            const float nrm = __shfl(nv, lanebase + k, 32);
            const float4 hv = ((const float4*)(h + (long)s * HID))[j];
            acc.x += hv.x * nrm;
            acc.y += hv.y * nrm;
            acc.z += hv.z * nrm;
            acc.w += hv.w * nrm;
        }
    }

    float4 r;
    r.x = fmaxf(acc.x, 0.0f);
    r.y = fmaxf(acc.y, 0.0f);
    r.z = fmaxf(acc.z, 0.0f);
    r.w = fmaxf(acc.w, 0.0f);
    ((float4*)(out + i * HID))[j] = r;
}

// ------------------------------------------------- pooling (batch is sorted)
// Contiguous node range per walker; register accumulator; atomic flush only on
// graph-boundary change -> ~50k atomics instead of 6.4M.
#define POOL_BLOCKS 32
__global__ __launch_bounds__(256) void pool_kernel(const int* __restrict__ batch,
                                                   const float* __restrict__ h,
                                                   float* __restrict__ pooled) {
    const int tid = threadIdx.x;
    const int w   = tid >> 4;       // walker 0..15
    const int j   = tid & 15;       // float4 slot
    const long per_block = (N_NODES + POOL_BLOCKS - 1) / POOL_BLOCKS;
    const long lo = (long)blockIdx.x * per_block;
    const long hi = (lo + per_block < N_NODES) ? lo + per_block : N_NODES;
    const long per_walk = (per_block + 15) / 16;
    const long s = lo + w * per_walk;
    const long e = (s + per_walk < hi) ? s + per_walk : hi;
    if (s >= e) return;

    int cur = batch[s];
    float4 acc = make_float4(0.f, 0.f, 0.f, 0.f);
    for (long i = s; i < e; ++i) {
        const int g = batch[i];
        if (g != cur) {
            float* pp = pooled + (long)cur * HID + j * 4;
            atomicAdd(pp + 0, acc.x); atomicAdd(pp + 1, acc.y);
            atomicAdd(pp + 2, acc.z); atomicAdd(pp + 3, acc.w);
            acc = make_float4(0.f, 0.f, 0.f, 0.f);
            cur = g;
        }
        const float4 hv = ((const float4*)(h + i * HID))[j];
        acc.x += hv.x; acc.y += hv.y; acc.z += hv.z; acc.w += hv.w;
    }
    float* pp = pooled + (long)cur * HID + j * 4;
    atomicAdd(pp + 0, acc.x); atomicAdd(pp + 1, acc.y);
    atomicAdd(pp + 2, acc.z); atomicAdd(pp + 3, acc.w);
}

// out[G,2] = concat(pooled, graph_feats) @ Wlin + blin   (tiny head)
__global__ void head_kernel(const float* __restrict__ pooled,
                            const float* __restrict__ gfeat,
                            const float* __restrict__ Wlin,
                            const float* __restrict__ blin,
                            float* __restrict__ out) {
    int g = threadIdx.x;
    if (g >= NGRAPH) return;
    float acc0 = blin[0], acc1 = blin[1];
    #pragma unroll 8
    for (int k = 0; k < HID; ++k) {
        const float p = pooled[g * HID + k];
        acc0 += p * Wlin[k * NOUT + 0];
        acc1 += p * Wlin[k * NOUT + 1];
    }
    const float gf = gfeat[g];
    acc0 += gf * Wlin[HID * NOUT + 0];
    acc1 += gf * Wlin[HID * NOUT + 1];
    out[g * NOUT + 0] = acc0;
    out[g * NOUT + 1] = acc1;
}

// ---------------------------------------------------------------- launcher

extern "C" void kernel_launch(void* const* d_in, const int* in_sizes, int n_in,
                              void* d_out, int out_size, void* d_ws, size_t ws_size,
                              hipStream_t stream) {
    const float* x     = (const float*)d_in[0];
    const int*   eidx  = (const int*)d_in[1];      // [2, E]: row 0 = src, row 1 = dst
    const int*   batch = (const int*)d_in[2];
    const float* gfeat = (const float*)d_in[3];
    const float* W1    = (const float*)d_in[4];
    const float* b1    = (const float*)d_in[5];
    const float* W2    = (const float*)d_in[6];
    const float* b2    = (const float*)d_in[7];
    const float* Wlin  = (const float*)d_in[8];
    const float* blin  = (const float*)d_in[9];
    float*       out   = (float*)d_out;

    const int* src = eidx;
    const int* dst = eidx + N_EDGES;

    // Workspace layout (16B-aligned slices; ~57 MB total)
    int*   count   = (int*)d_ws;                        // N
    int*   offsets = count + N_NODES;                   // N+16 (padded)
    int*   cursor  = offsets + N_NODES + 16;            // N
    int*   csr_src = cursor + N_NODES;                  // E
    float* dinv    = (float*)(csr_src + N_EDGES);       // N
    float* bufA    = dinv + N_NODES;                    // N*H  (h = in @ W)
    float* bufC    = bufA + (long)N_NODES * HID;        // N*H  (layer out)
    float* pooled  = bufC + (long)N_NODES * HID;        // G*H

    const int T = 256;
    const int gN    = (N_NODES + T - 1) / T;
    const int gE    = (N_EDGES + T - 1) / T;
    const int gN16  = (int)(((long)N_NODES * 16 + T - 1) / T);   // 6250
    const int gGemm = N_NODES / 16;                               // 6250, exact

    // ---- CSR build (once; reused by both layers) + dinv
    zero_kernel<<<gN, T, 0, stream>>>((float*)count, N_NODES);   // int 0 == 0.0f bits
    count_kernel<<<gE, T, 0, stream>>>(dst, count);
    scan_kernel<<<1, 1024, 0, stream>>>(count, offsets, cursor, dinv);
    fill_kernel<<<gE, T, 0, stream>>>(src, dst, cursor, csr_src);

    // ---- layer 1: GEMM (WMMA f32) + fused gather/self/bias/relu (no atomics)
    gemm_wmma_kernel<<<gGemm, 128, 0, stream>>>(x, W1, bufA);
    gather_kernel<<<gN16, T, 0, stream>>>(offsets, csr_src, dinv, bufA, b1, bufC);

    // ---- layer 2
    gemm_wmma_kernel<<<gGemm, 128, 0, stream>>>(bufC, W2, bufA);
    gather_kernel<<<gN16, T, 0, stream>>>(offsets, csr_src, dinv, bufA, b2, bufC);

    // ---- pool + head
    zero_kernel<<<16, T, 0, stream>>>(pooled, (long)NGRAPH * HID);
    pool_kernel<<<POOL_BLOCKS, T, 0, stream>>>(batch, bufC, pooled);
    head_kernel<<<1, 64, 0, stream>>>(pooled, gfeat, Wlin, blin, out);
}